// bertCSRModel_2860448219829
// MI455X (gfx1250) — compile-verified
//
#include <hip/hip_runtime.h>
#include <hip/hip_bf16.h>

// ---------------------------------------------------------------------------
// Model constants (from the reference): n_sents=64, S=512, D=768,
// E=2000, P=20000, NL=128, H=512, C=3, L_SPAN=8, MAX_ITER=3, RATE=0.3
// ---------------------------------------------------------------------------
#define DD     768
#define SS     512
#define NL     128
#define HH     512
#define RATE   0.3f
#define BETA_INTRA 0.7f
#define BETA_INTER 0.3f
#define KC     32          // K-chunk staged in LDS (multiple of 4; divides 768 & 128)

typedef float v2f __attribute__((ext_vector_type(2)));
typedef float v8f __attribute__((ext_vector_type(8)));

// ---------------------------------------------------------------------------
// FP32 WMMA GEMM:  C = alpha * A(MxK) @ B(KxN)  [+ C if accum] [relu if flag]
// Row-major. Requires M%16==0, N%64==0, K%KC==0 (true for all call sites).
// Block = 128 threads (4 waves) -> one 16x64 strip of C; each wave owns one
// 16x16 tile via v_wmma_f32_16x16x4_f32.
// The 16xKC A strip (shared by all 4 waves) is staged in LDS with coalesced
// float4 loads, double-buffered so the next chunk's global load overlaps the
// current chunk's 8 WMMAs. Row pitch KC+1 keeps 16-lane column reads
// conflict-free across the 64 LDS banks.
// ---------------------------------------------------------------------------
__global__ __launch_bounds__(128)
void wmma_gemm_f32(const float* __restrict__ A, int lda,
                   const float* __restrict__ B, int ldb,
                   float* __restrict__ C, int ldc,
                   int M, int N, int K,
                   float alpha, int accum, int do_relu)
{
    __shared__ float As[2][16][KC + 1];

    const int tid  = threadIdx.x;
    const int lane = tid & 31;
    const int wave = tid >> 5;
    const int m0 = blockIdx.y << 4;
    const int n0 = (blockIdx.x * 4 + wave) << 4;

    const int r  = lane & 15;            // row within A tile / col within B tile
    const int kh = (lane >> 4) << 1;     // lanes 0-15 -> K offset 0, lanes 16-31 -> 2

    // Staging: 16 rows x KC cols = 128 float4's -> one per thread, coalesced.
    const int srow = tid >> 3;           // 0..15
    const int scol = (tid & 7) << 2;     // 0,4,...,28
    const float* Ag = A + (size_t)(m0 + srow) * lda + scol;

    const float* Bcol = B + (size_t)kh * ldb + (n0 + r);

    const int nchunks = K / KC;

    // Stage chunk 0
    {
        const float4 v = *(const float4*)Ag;
        float* dst = &As[0][srow][scol];
        dst[0] = v.x; dst[1] = v.y; dst[2] = v.z; dst[3] = v.w;
    }
    __syncthreads();

    v8f acc = {};
    for (int c = 0; c < nchunks; ++c) {
        // Stage next chunk into the other buffer while computing this one.
        if (c + 1 < nchunks) {
            const float4 v = *(const float4*)(Ag + (size_t)(c + 1) * KC);
            float* dst = &As[(c + 1) & 1][srow][scol];
            dst[0] = v.x; dst[1] = v.y; dst[2] = v.z; dst[3] = v.w;
            if (c + 2 < nchunks)
                __builtin_prefetch(Ag + (size_t)(c + 2) * KC, 0, 1);  // global_prefetch_b8
        }

        const float (*Asb)[KC + 1] = As[c & 1];
        const int kbase = c * KC;
#pragma unroll
        for (int kc = 0; kc < KC; kc += 4) {
            v2f a, b;
            a.x = Asb[r][kc + kh];
            a.y = Asb[r][kc + kh + 1];
            const size_t kg = (size_t)(kbase + kc);
            b.x = Bcol[kg * ldb];
            b.y = Bcol[(kg + 1) * ldb];
            acc = __builtin_amdgcn_wmma_f32_16x16x4_f32(
                      false, a, false, b, (short)0, acc, false, false);
        }
        __syncthreads();
    }

    // C/D layout: vgpr g -> M = m0 + g + (lane>=16 ? 8 : 0), N = n0 + (lane&15)
    const int mb = m0 + ((lane >> 4) << 3);
    const int nc = n0 + r;
#pragma unroll
    for (int g = 0; g < 8; ++g) {
        size_t idx = (size_t)(mb + g) * ldc + nc;
        float v = alpha * acc[g];
        if (accum)   v += C[idx];
        if (do_relu) v = fmaxf(v, 0.0f);
        C[idx] = v;
    }
}

// ---------------------------------------------------------------------------
// Event span extraction: e_emb[e] = mean of tokens [start, start+len] (masked)
// One block per event, 256 threads, 3 dims each (768 total).
// ---------------------------------------------------------------------------
__global__ __launch_bounds__(256)
void extract_events_kernel(const float* __restrict__ sent_emb,
                           const int* __restrict__ ev_sent,
                           const int* __restrict__ ev_start,
                           const int* __restrict__ ev_len,
                           float* __restrict__ e_emb)
{
    const int e  = blockIdx.x;
    const int d  = threadIdx.x;
    const int s  = ev_sent[e];
    const int st = ev_start[e];
    const int ln = ev_len[e] + 1;              // 1..8
    const float inv = 1.0f / (float)ln;
    const float* base = sent_emb + ((size_t)s * SS + st) * DD;
#pragma unroll
    for (int j = 0; j < 3; ++j) {
        const int dd = d + j * 256;
        float acc = 0.0f;
        for (int t = 0; t < ln; ++t)
            acc += base[(size_t)t * DD + dd];
        e_emb[(size_t)e * DD + dd] = acc * inv;
    }
}

// ---------------------------------------------------------------------------
// Per-iteration aggregation init: s_intra = x, c_intra = 1, s_inter/c_inter = 0
// ---------------------------------------------------------------------------
__global__ __launch_bounds__(256)
void init_agg_kernel(const float* __restrict__ x,
                     float* __restrict__ s_intra, float* __restrict__ c_intra,
                     float* __restrict__ s_inter, float* __restrict__ c_inter,
                     int E)
{
    const int i = blockIdx.x * 256 + threadIdx.x;
    if (i < E * NL) {
        s_intra[i] = x[i];
        s_inter[i] = 0.0f;
    }
    if (i < E) {
        c_intra[i] = 1.0f;
        c_inter[i] = 0.0f;
    }
}

__global__ void zero_loss_kernel(float* loss) { if (threadIdx.x == 0) loss[0] = 0.0f; }

// ---------------------------------------------------------------------------
// Fused pair kernel: one block (128 threads) per pair.
//   h      = relu(A1[p1] + A2[p2] + Xd[p1] - Xd[p2] + b1)        (H=512)
//   logits = h @ W2 + b2                                          (C=3)
//   loss  += -log_softmax(logits)[target] * inv_scale             (atomic)
//   masks -> segment-sum atomics on s_intra/s_inter/c_intra/c_inter
// ---------------------------------------------------------------------------
__global__ __launch_bounds__(128)
void pair_forward_kernel(const float* __restrict__ A1, const float* __restrict__ A2,
                         const float* __restrict__ Xd,
                         const float* __restrict__ b1,
                         const float* __restrict__ W2, const float* __restrict__ b2,
                         const float* __restrict__ x,
                         const int* __restrict__ pair1, const int* __restrict__ pair2,
                         const int* __restrict__ rel_type, const int* __restrict__ target,
                         float* __restrict__ logits_out, float* __restrict__ loss_out,
                         float* __restrict__ s_intra, float* __restrict__ c_intra,
                         float* __restrict__ s_inter, float* __restrict__ c_inter,
                         float inv_scale)
{
    const int p = blockIdx.x;
    const int t = threadIdx.x;                 // 0..127
    const int i1 = pair1[p];
    const int i2 = pair2[p];

    const float* a1  = A1 + (size_t)i1 * HH;
    const float* a2  = A2 + (size_t)i2 * HH;
    const float* xd1 = Xd + (size_t)i1 * HH;
    const float* xd2 = Xd + (size_t)i2 * HH;

    float l0 = 0.0f, l1 = 0.0f, l2 = 0.0f;
#pragma unroll
    for (int j = t; j < HH; j += 128) {
        float h = a1[j] + a2[j] + xd1[j] - xd2[j] + b1[j];
        h = fmaxf(h, 0.0f);
        l0 = fmaf(h, W2[j * 3 + 0], l0);
        l1 = fmaf(h, W2[j * 3 + 1], l1);
        l2 = fmaf(h, W2[j * 3 + 2], l2);
    }

    __shared__ float red[3 * 128];
    __shared__ float flags[4];
    red[t] = l0; red[128 + t] = l1; red[256 + t] = l2;
    __syncthreads();
    for (int s = 64; s > 0; s >>= 1) {
        if (t < s) {
            red[t]       += red[t + s];
            red[128 + t] += red[128 + t + s];
            red[256 + t] += red[256 + t + s];
        }
        __syncthreads();
    }

    if (t == 0) {
        const float L0 = red[0]   + b2[0];
        const float L1 = red[128] + b2[1];
        const float L2 = red[256] + b2[2];
        logits_out[(size_t)p * 3 + 0] = L0;
        logits_out[(size_t)p * 3 + 1] = L1;
        logits_out[(size_t)p * 3 + 2] = L2;

        const float mx = fmaxf(L0, fmaxf(L1, L2));
        const float e0 = expf(L0 - mx), e1 = expf(L1 - mx), e2 = expf(L2 - mx);
        const float sum = e0 + e1 + e2;
        const float lse = mx + logf(sum);

        const int tg = target[p];
        const float Lt = (tg == 0) ? L0 : (tg == 1) ? L1 : L2;
        atomicAdd(loss_out, (lse - Lt) * inv_scale);   // -logp[target] * inv_scale

        // softmax probs, argmax (first-max semantics like jnp.argmax)
        const float p0 = e0 / sum, p1 = e1 / sum, p2 = e2 / sum;
        int   argm = 0;
        float pm   = p0;
        if (p1 > pm) { pm = p1; argm = 1; }
        if (p2 > pm) { pm = p2; argm = 2; }
        const bool conf = pm > RATE;
        const int  rt   = rel_type[p];

        const float mi1 = (conf && argm == 1 && rt == 0) ? 1.0f : 0.0f;
        const float mi2 = (conf && argm == 2 && rt == 0) ? 1.0f : 0.0f;
        const float me1 = (conf && argm == 1 && rt == 1) ? 1.0f : 0.0f;
        const float me2 = (conf && argm == 2 && rt == 1) ? 1.0f : 0.0f;
        flags[0] = mi1; flags[1] = mi2; flags[2] = me1; flags[3] = me2;
        if (mi1 != 0.0f) atomicAdd(&c_intra[i2], 1.0f);
        if (mi2 != 0.0f) atomicAdd(&c_intra[i1], 1.0f);
        if (me1 != 0.0f) atomicAdd(&c_inter[i2], 1.0f);
        if (me2 != 0.0f) atomicAdd(&c_inter[i1], 1.0f);
    }
    __syncthreads();

    // 128 threads = NL dims: masked segment-sum scatter of x rows
    const float x1 = x[(size_t)i1 * NL + t];
    const float x2 = x[(size_t)i2 * NL + t];
    if (flags[0] != 0.0f) atomicAdd(&s_intra[(size_t)i2 * NL + t], x1);
    if (flags[1] != 0.0f) atomicAdd(&s_intra[(size_t)i1 * NL + t], x2);
    if (flags[2] != 0.0f) atomicAdd(&s_inter[(size_t)i2 * NL + t], x1);
    if (flags[3] != 0.0f) atomicAdd(&s_inter[(size_t)i1 * NL + t], x2);
}

// ---------------------------------------------------------------------------
// agg_intra = s_intra / c_intra ; agg_inter = s_inter / max(c_inter, 1)
// ---------------------------------------------------------------------------
__global__ __launch_bounds__(256)
void finalize_agg_kernel(const float* __restrict__ s_intra, const float* __restrict__ c_intra,
                         const float* __restrict__ s_inter, const float* __restrict__ c_inter,
                         float* __restrict__ agg_intra, float* __restrict__ agg_inter,
                         int E)
{
    const int i = blockIdx.x * 256 + threadIdx.x;
    if (i >= E * NL) return;
    const int e = i >> 7;   // /NL
    agg_intra[i] = s_intra[i] / c_intra[e];
    agg_inter[i] = s_inter[i] / fmaxf(c_inter[e], 1.0f);
}

// ---------------------------------------------------------------------------
// Host-side orchestration
// ---------------------------------------------------------------------------
extern "C" void kernel_launch(void* const* d_in, const int* in_sizes, int n_in,
                              void* d_out, int out_size, void* d_ws, size_t ws_size,
                              hipStream_t stream)
{
    const float* sent_emb = (const float*)d_in[0];   // (64, 512, 768)
    const float* W_proj   = (const float*)d_in[1];   // (768, 128)
    const float* W1       = (const float*)d_in[2];   // (1664, 512)
    const float* b1       = (const float*)d_in[3];   // (512,)
    const float* W2       = (const float*)d_in[4];   // (512, 3)
    const float* b2       = (const float*)d_in[5];   // (3,)
    const float* W_self   = (const float*)d_in[6];   // (128, 128)
    const float* W_intra  = (const float*)d_in[7];   // (128, 128)
    const float* W_inter  = (const float*)d_in[8];   // (128, 128)
    const int* ev_sent    = (const int*)d_in[9];
    const int* ev_start   = (const int*)d_in[10];
    const int* ev_len     = (const int*)d_in[11];
    const int* pair1      = (const int*)d_in[12];
    const int* pair2      = (const int*)d_in[13];
    const int* rel_type   = (const int*)d_in[14];
    const int* target     = (const int*)d_in[15];

    const int E = in_sizes[9];    // 2000 (multiple of 16)
    const int P = in_sizes[12];   // 20000

    // Workspace carve-out (~24.6 MB of fp32)
    float* ws = (float*)d_ws;
    float* e_emb     = ws;  ws += (size_t)E * DD;   // E x 768
    float* A1        = ws;  ws += (size_t)E * HH;   // E x 512
    float* A2        = ws;  ws += (size_t)E * HH;
    float* Xd        = ws;  ws += (size_t)E * HH;
    float* xA        = ws;  ws += (size_t)E * NL;   // ping
    float* xB        = ws;  ws += (size_t)E * NL;   // pong
    float* s_intra   = ws;  ws += (size_t)E * NL;
    float* s_inter   = ws;  ws += (size_t)E * NL;
    float* agg_intra = ws;  ws += (size_t)E * NL;
    float* agg_inter = ws;  ws += (size_t)E * NL;
    float* c_intra   = ws;  ws += E;
    float* c_inter   = ws;  ws += E;

    float* loss   = (float*)d_out;          // scalar
    float* logits = (float*)d_out + 1;      // P x 3

    zero_loss_kernel<<<1, 32, 0, stream>>>(loss);

    // e_emb (E x 768)
    extract_events_kernel<<<E, 256, 0, stream>>>(sent_emb, ev_sent, ev_start, ev_len, e_emb);

    // One-time GEMMs (feats @ W1 factorization + projection)
    // A1 = e_emb @ W1[0:768, :]        (E x 512)
    wmma_gemm_f32<<<dim3(HH / 64, E / 16), 128, 0, stream>>>(
        e_emb, DD, W1, HH, A1, HH, E, HH, DD, 1.0f, 0, 0);
    // A2 = e_emb @ W1[768:1536, :]     (E x 512)
    wmma_gemm_f32<<<dim3(HH / 64, E / 16), 128, 0, stream>>>(
        e_emb, DD, W1 + (size_t)DD * HH, HH, A2, HH, E, HH, DD, 1.0f, 0, 0);
    // x0 = e_emb @ W_proj              (E x 128)
    wmma_gemm_f32<<<dim3(NL / 64, E / 16), 128, 0, stream>>>(
        e_emb, DD, W_proj, NL, xA, NL, E, NL, DD, 1.0f, 0, 0);

    float* xcur = xA;
    float* xnext = xB;
    const int aggBlocks = (E * NL + 255) / 256;

    for (int it = 0; it < 3; ++it) {
        // Xd = x @ W1[1536:1664, :]    (E x 512)
        wmma_gemm_f32<<<dim3(HH / 64, E / 16), 128, 0, stream>>>(
            xcur, NL, W1 + (size_t)(2 * DD) * HH, HH, Xd, HH, E, HH, NL, 1.0f, 0, 0);

        init_agg_kernel<<<aggBlocks, 256, 0, stream>>>(xcur, s_intra, c_intra, s_inter, c_inter, E);

        pair_forward_kernel<<<P, 128, 0, stream>>>(
            A1, A2, Xd, b1, W2, b2, xcur, pair1, pair2, rel_type, target,
            logits, loss, s_intra, c_intra, s_inter, c_inter,
            1.0f / ((float)P * (float)(it + 1)));

        finalize_agg_kernel<<<aggBlocks, 256, 0, stream>>>(
            s_intra, c_intra, s_inter, c_inter, agg_intra, agg_inter, E);

        // x_next = relu(x @ W_self + 0.7*(agg_intra @ W_intra) + 0.3*(agg_inter @ W_inter))
        wmma_gemm_f32<<<dim3(NL / 64, E / 16), 128, 0, stream>>>(
            xcur, NL, W_self, NL, xnext, NL, E, NL, NL, 1.0f, 0, 0);
        wmma_gemm_f32<<<dim3(NL / 64, E / 16), 128, 0, stream>>>(
            agg_intra, NL, W_intra, NL, xnext, NL, E, NL, NL, BETA_INTRA, 1, 0);
        wmma_gemm_f32<<<dim3(NL / 64, E / 16), 128, 0, stream>>>(
            agg_inter, NL, W_inter, NL, xnext, NL, E, NL, NL, BETA_INTER, 1, 1);

        float* tmp = xcur; xcur = xnext; xnext = tmp;
    }
}